// Speller_53644141527187
// MI455X (gfx1250) — compile-verified
//
#include <hip/hip_runtime.h>
#include <math.h>

typedef __attribute__((ext_vector_type(2))) float v2f;
typedef __attribute__((ext_vector_type(4))) float v4f;
typedef __attribute__((ext_vector_type(8))) float v8f;

#define BB 128
#define TT 2048
#define OO 128
#define EE 512
#define LL 34
#define KX  (EE + OO)     // 640
#define G4  (4 * OO)      // 512
#define CH  4
#define TC  (TT / CH)     // 512

// d_out float offsets (return order: out, c, sh, sc)
#define OFF_OUT 0
#define OFF_C   (BB * LL)               // 4352
#define OFF_SH  (OFF_C + BB * OO)       // 20736
#define OFF_SC  (OFF_SH + BB * OO)      // 37120

// d_ws float offsets
#define WS_GATES 0                       // 128*512
#define WS_SCORE (WS_GATES + BB * G4)    // 128*2048
#define WS_PMAX  (WS_SCORE + BB * TT)    // 512
#define WS_FMAX  (WS_PMAX + BB * CH)     // 128
#define WS_PSUM  (WS_FMAX + BB)          // 512
#define WS_PC    (WS_PSUM + BB * CH)     // 512*128

// ---------------------------------------------------------------------------
// Kernel 1: gates = [y1|c1|sh1] @ [W_ih|W_hh]^T + b_ih + b_hh  via f32 WMMA.
// One wave per 16x16 tile of gates[128, 512]. K split into three statically
// bounded loops so address selection is uniform (no EXEC-divergent cndmask).
// ---------------------------------------------------------------------------
__global__ __launch_bounds__(32)
void gates_wmma(const float* __restrict__ y1, const float* __restrict__ c1,
                const float* __restrict__ sh1,
                const float* __restrict__ Wih, const float* __restrict__ bih,
                const float* __restrict__ Whh, const float* __restrict__ bhh,
                float* __restrict__ gates)
{
    const int tile = blockIdx.x;       // 0..255
    const int tn = tile & 31;          // N tile (512/16)
    const int tm = tile >> 5;          // M tile (128/16)
    const int lane = threadIdx.x;      // 0..31
    const int lmn  = lane & 15;
    const int hi   = lane >> 4;        // 0 or 1
    const int koff = hi * 2;

    const int m = tm * 16 + lmn;       // batch row carried by this lane (A)
    const int n = tn * 16 + lmn;       // gate column carried by this lane (B)

    // fixed per-lane base pointers; loop index contributes constant offsets
    const float* aY = y1  + (size_t)m * EE + koff;
    const float* aC = c1  + (size_t)m * OO + koff;
    const float* aH = sh1 + (size_t)m * OO + koff;
    const float* bW = Wih + (size_t)n * KX + koff;
    const float* bH = Whh + (size_t)n * OO + koff;

    v8f acc = {};
    // segment 1: K = 0..511  (x = y1, W = Wih)
#pragma unroll 8
    for (int k = 0; k < EE; k += 4) {
        v2f a  = *(const v2f*)(aY + k);
        v2f bv = *(const v2f*)(bW + k);
        acc = __builtin_amdgcn_wmma_f32_16x16x4_f32(false, a, false, bv,
                                                    (short)0, acc, false, false);
    }
    // segment 2: K = 512..639  (x = c1, W = Wih[:,512:])
#pragma unroll 8
    for (int k = 0; k < OO; k += 4) {
        v2f a  = *(const v2f*)(aC + k);
        v2f bv = *(const v2f*)(bW + EE + k);
        acc = __builtin_amdgcn_wmma_f32_16x16x4_f32(false, a, false, bv,
                                                    (short)0, acc, false, false);
    }
    // segment 3: K = 640..767  (x = sh1, W = Whh)
#pragma unroll 8
    for (int k = 0; k < OO; k += 4) {
        v2f a  = *(const v2f*)(aH + k);
        v2f bv = *(const v2f*)(bH + k);
        acc = __builtin_amdgcn_wmma_f32_16x16x4_f32(false, a, false, bv,
                                                    (short)0, acc, false, false);
    }

    // C/D layout: VGPR r -> lanes 0-15: (M=r, N=lane); lanes 16-31: (M=8+r, N=lane-16)
    const float bias = bih[n] + bhh[n];
    const int rowbase = tm * 16 + hi * 8;
#pragma unroll
    for (int r = 0; r < 8; ++r)
        gates[(size_t)(rowbase + r) * G4 + n] = acc[r] + bias;
}

// ---------------------------------------------------------------------------
// Kernel 2: LSTM elementwise -> sc, sh (PyTorch gate order i,f,g,o)
// ---------------------------------------------------------------------------
__global__ __launch_bounds__(256)
void lstm_cell(const float* __restrict__ gates, const float* __restrict__ sc1,
               float* __restrict__ sh_out, float* __restrict__ sc_out)
{
    const int idx = blockIdx.x * 256 + threadIdx.x;  // 0..16383
    const int b = idx >> 7;
    const int o = idx & (OO - 1);
    const float* g = gates + (size_t)b * G4;
    const float gi = g[o], gf = g[o + OO], gg = g[o + 2 * OO], go = g[o + 3 * OO];
    const float si = 1.f / (1.f + __expf(-gi));
    const float sf = 1.f / (1.f + __expf(-gf));
    const float so = 1.f / (1.f + __expf(-go));
    const float sc = sf * sc1[idx] + si * tanhf(gg);
    sc_out[idx] = sc;
    sh_out[idx] = so * tanhf(sc);
}

// ---------------------------------------------------------------------------
// Kernel 3a: scores[b,t] = hk[b,t,:]·sc[b,:] via f32 WMMA (sc broadcast into
// all 16 B columns); one WG per (b, T-chunk); also per-chunk masked max.
// hk is 134 MB streamed once -> non-temporal loads (don't rinse L2).
// ---------------------------------------------------------------------------
__global__ __launch_bounds__(256)
void attn_scores(const float* __restrict__ hk, const float* __restrict__ mask,
                 const float* __restrict__ sc, float* __restrict__ scorew,
                 float* __restrict__ pmax)
{
    __shared__ float s_sc[OO];
    __shared__ float sscore[TC];
    __shared__ float red[256];
    const int blk = blockIdx.x;            // 0..511
    const int b   = blk >> 2;
    const int ch  = blk & (CH - 1);
    const int tid = threadIdx.x;
    const int lane = tid & 31;
    const int wave = tid >> 5;
    if (tid < OO) s_sc[tid] = sc[b * OO + tid];
    __syncthreads();

    const int lmn  = lane & 15;
    const int hi   = lane >> 4;
    const int koff = hi * 2;
    const int tbase = ch * TC;
    const float* hkb = hk + ((size_t)b * TT + tbase) * OO + koff;

    for (int tile = wave; tile < TC / 16; tile += 8) {
        const int t0 = tile * 16;
        const float* rowp = hkb + (size_t)(t0 + lmn) * OO;
        v8f acc = {};
#pragma unroll 8
        for (int k = 0; k < OO; k += 4) {
            v2f a = __builtin_nontemporal_load((const v2f*)(rowp + k));
            v2f bv;
            bv.x = s_sc[k + koff];
            bv.y = s_sc[k + koff + 1];
            acc = __builtin_amdgcn_wmma_f32_16x16x4_f32(false, a, false, bv,
                                                        (short)0, acc, false, false);
        }
        if (lmn == 0) {                    // all N columns equal; lanes 0/16 extract
#pragma unroll
            for (int r = 0; r < 8; ++r)
                sscore[t0 + hi * 8 + r] = acc[r];
        }
    }
    __syncthreads();

    float vmax = -INFINITY;
    for (int t = tid; t < TC; t += 256) {
        const float sv = sscore[t];
        scorew[(size_t)b * TT + tbase + t] = sv;
        if (mask[(size_t)b * TT + tbase + t] > 0.5f) vmax = fmaxf(vmax, sv);
    }
    red[tid] = vmax;
    __syncthreads();
    for (int s = 128; s > 0; s >>= 1) {
        if (tid < s) red[tid] = fmaxf(red[tid], red[tid + s]);
        __syncthreads();
    }
    if (tid == 0) pmax[blk] = red[0];
}

// Kernel 3b: final masked max per batch
__global__ __launch_bounds__(128)
void attn_max(const float* __restrict__ pmax, float* __restrict__ fmax)
{
    const int b = threadIdx.x;
    float m = pmax[b * CH];
    for (int ch = 1; ch < CH; ++ch) m = fmaxf(m, pmax[b * CH + ch]);
    fmax[b] = m;
}

// ---------------------------------------------------------------------------
// Kernel 3c: per-chunk exp-weights + partial exp-sum + partial weighted sum
// of hv rows (float4-coalesced non-temporal streaming of hv, read once).
// ---------------------------------------------------------------------------
__global__ __launch_bounds__(256)
void attn_apply(const float* __restrict__ hv, const float* __restrict__ mask,
                const float* __restrict__ scorew, const float* __restrict__ fmax,
                float* __restrict__ psum, float* __restrict__ pc)
{
    __shared__ float escore[TC];
    __shared__ float red[256];
    __shared__ v4f cred4[8 * (OO / 4)];
    const int blk = blockIdx.x;
    const int b   = blk >> 2;
    const int ch  = blk & (CH - 1);
    const int tid = threadIdx.x;
    const int tbase = ch * TC;
    const float mx = fmax[b];

    float lsum = 0.f;
    for (int t = tid; t < TC; t += 256) {
        const float e = mask[(size_t)b * TT + tbase + t] *
                        __expf(scorew[(size_t)b * TT + tbase + t] - mx);
        escore[t] = e;
        lsum += e;
    }
    red[tid] = lsum;
    __syncthreads();
    for (int s = 128; s > 0; s >>= 1) {
        if (tid < s) red[tid] += red[tid + s];
        __syncthreads();
    }
    if (tid == 0) psum[blk] = red[0];

    const int g = tid >> 5;                 // row group 0..7
    const int q = tid & 31;                 // v4f slot within 128-wide row
    const v4f* hv4 = (const v4f*)(hv + ((size_t)b * TT + tbase) * OO);
    v4f acc4 = {};
#pragma unroll 4
    for (int t = g; t < TC; t += 8) {
        const float w = escore[t];
        const v4f v = __builtin_nontemporal_load(hv4 + (size_t)t * (OO / 4) + q);
        acc4 += w * v;
    }
    cred4[g * (OO / 4) + q] = acc4;
    __syncthreads();
    if (tid < OO) {
        const float* cr = (const float*)cred4;
        float s = 0.f;
#pragma unroll
        for (int gg = 0; gg < 8; ++gg) s += cr[gg * OO + tid];
        pc[(size_t)blk * OO + tid] = s;
    }
}

// Kernel 3d: combine chunk partials -> normalized context c
__global__ __launch_bounds__(128)
void attn_final(const float* __restrict__ psum, const float* __restrict__ pc,
                float* __restrict__ c_out)
{
    const int b = blockIdx.x;
    const int o = threadIdx.x;
    float s = 0.f, cv = 0.f;
    for (int ch = 0; ch < CH; ++ch) {
        s  += psum[b * CH + ch];
        cv += pc[(size_t)(b * CH + ch) * OO + o];
    }
    c_out[(size_t)b * OO + o] = cv / s;
}

// ---------------------------------------------------------------------------
// Kernel 4: out = softmax(relu(sh@W1^T + b1 + c@W2^T + b2) @ W3^T + b3)
// ---------------------------------------------------------------------------
__global__ __launch_bounds__(64)
void out_mlp(const float* __restrict__ sh, const float* __restrict__ c,
             const float* __restrict__ W1, const float* __restrict__ b1,
             const float* __restrict__ W2, const float* __restrict__ b2,
             const float* __restrict__ W3, const float* __restrict__ b3,
             float* __restrict__ out)
{
    __shared__ float h[LL];
    __shared__ float z[LL];
    __shared__ float smx, ssum;
    const int b = blockIdx.x;
    const int l = threadIdx.x;
    if (l < LL) {
        float s = b1[l] + b2[l];
        const float* w1 = W1 + (size_t)l * OO;
        const float* w2 = W2 + (size_t)l * OO;
        const float* shb = sh + (size_t)b * OO;
        const float* cb  = c  + (size_t)b * OO;
        for (int o = 0; o < OO; ++o) s += shb[o] * w1[o] + cb[o] * w2[o];
        h[l] = fmaxf(s, 0.f);
    }
    __syncthreads();
    if (l < LL) {
        float s = b3[l];
        const float* w3 = W3 + (size_t)l * LL;
        for (int j = 0; j < LL; ++j) s += h[j] * w3[j];
        z[l] = s;
    }
    __syncthreads();
    if (l == 0) {
        float m = z[0];
        for (int j = 1; j < LL; ++j) m = fmaxf(m, z[j]);
        smx = m;
    }
    __syncthreads();
    if (l < LL) z[l] = __expf(z[l] - smx);
    __syncthreads();
    if (l == 0) {
        float s = 0.f;
        for (int j = 0; j < LL; ++j) s += z[j];
        ssum = s;
    }
    __syncthreads();
    if (l < LL) out[(size_t)b * LL + l] = z[l] / ssum;
}

// ---------------------------------------------------------------------------
extern "C" void kernel_launch(void* const* d_in, const int* in_sizes, int n_in,
                              void* d_out, int out_size, void* d_ws, size_t ws_size,
                              hipStream_t stream)
{
    (void)in_sizes; (void)n_in; (void)out_size; (void)ws_size;
    const float* hk   = (const float*)d_in[0];
    const float* hv   = (const float*)d_in[1];
    const float* y1   = (const float*)d_in[2];
    const float* c1   = (const float*)d_in[3];
    const float* sh1  = (const float*)d_in[4];
    const float* sc1  = (const float*)d_in[5];
    const float* mask = (const float*)d_in[6];
    const float* Wih  = (const float*)d_in[7];
    const float* bih  = (const float*)d_in[8];
    const float* Whh  = (const float*)d_in[9];
    const float* bhh  = (const float*)d_in[10];
    const float* W1   = (const float*)d_in[11];
    const float* b1   = (const float*)d_in[12];
    const float* W2   = (const float*)d_in[13];
    const float* b2   = (const float*)d_in[14];
    const float* W3   = (const float*)d_in[15];
    const float* b3   = (const float*)d_in[16];

    float* out  = (float*)d_out;
    float* c_o  = out + OFF_C;
    float* sh_o = out + OFF_SH;
    float* sc_o = out + OFF_SC;

    float* ws     = (float*)d_ws;
    float* gates  = ws + WS_GATES;
    float* scorew = ws + WS_SCORE;
    float* pmax   = ws + WS_PMAX;
    float* fmaxw  = ws + WS_FMAX;
    float* psum   = ws + WS_PSUM;
    float* pc     = ws + WS_PC;

    gates_wmma <<<dim3(256),    dim3(32),  0, stream>>>(y1, c1, sh1, Wih, bih, Whh, bhh, gates);
    lstm_cell  <<<dim3(64),     dim3(256), 0, stream>>>(gates, sc1, sh_o, sc_o);
    attn_scores<<<dim3(BB*CH),  dim3(256), 0, stream>>>(hk, mask, sc_o, scorew, pmax);
    attn_max   <<<dim3(1),      dim3(128), 0, stream>>>(pmax, fmaxw);
    attn_apply <<<dim3(BB*CH),  dim3(256), 0, stream>>>(hv, mask, scorew, fmaxw, psum, pc);
    attn_final <<<dim3(BB),     dim3(128), 0, stream>>>(psum, pc, c_o);
    out_mlp    <<<dim3(BB),     dim3(64),  0, stream>>>(sh_o, c_o, W1, b1, W2, b2, W3, b3, out);
}